// SSDLoss_20504173871767
// MI455X (gfx1250) — compile-verified
//
#include <hip/hip_runtime.h>
#include <math.h>

// Problem constants (match reference setup_inputs()).
constexpr int B    = 128;
constexpr int C    = 4;
constexpr int NBOX = 131072;

constexpr int SPLIT = 16;                // segments per row
constexpr int SEG   = NBOX / SPLIT;      // 8192 floats per segment
constexpr int T1    = 256;               // threads in scan kernel (8 wave32s)
constexpr int VPT   = SEG / (T1 * 4);    // float4 loads per thread = 8

typedef float v2f __attribute__((ext_vector_type(2)));
typedef float v4f __attribute__((ext_vector_type(4)));
typedef float v8f __attribute__((ext_vector_type(8)));

// Full total order matching jax.lax.top_k: larger value wins, ties -> lower idx.
__device__ __forceinline__ bool better(float av, int ai, float bv, int bi) {
    return (av > bv) || (av == bv && ai < bi);
}

// Branchless insert for the hot loop. PRECONDITION: the new element's index is
// strictly greater than every index already held (per-thread scan order), so a
// tie must keep the existing entry -> strict > compares suffice.
__device__ __forceinline__ void ins3_inc(float v, int i,
                                         float& v0, int& i0,
                                         float& v1, int& i1,
                                         float& v2, int& i2) {
    const bool b0 = v > v0;
    const bool b1 = v > v1;
    const bool b2 = v > v2;
    v2 = b1 ? v1 : (b2 ? v : v2);
    i2 = b1 ? i1 : (b2 ? i : i2);
    v1 = b0 ? v0 : (b1 ? v : v1);
    i1 = b0 ? i0 : (b1 ? i : i1);
    v0 = b0 ? v  : v0;
    i0 = b0 ? i  : i0;
}

// Branchless insert with the full (value, index) order, for merge stages where
// indices interleave arbitrarily.
__device__ __forceinline__ void ins3_full(float v, int i,
                                          float& v0, int& i0,
                                          float& v1, int& i1,
                                          float& v2, int& i2) {
    const bool b0 = better(v, i, v0, i0);
    const bool b1 = better(v, i, v1, i1);
    const bool b2 = better(v, i, v2, i2);
    v2 = b1 ? v1 : (b2 ? v : v2);
    i2 = b1 ? i1 : (b2 ? i : i2);
    v1 = b0 ? v0 : (b1 ? v : v1);
    i1 = b0 ? i0 : (b1 ? i : i1);
    v0 = b0 ? v  : v0;
    i0 = b0 ? i  : i0;
}

// Exact reference BCE-with-logits term.
__device__ __forceinline__ float bce(float x, float t) {
    return fmaxf(x, 0.0f) - x * t + log1pf(expf(-fabsf(x)));
}

// ---------------------------------------------------------------------------
// Kernel 1: per-(row,segment) top-3 of logits, excluding the positive anchor.
// Non-temporal b128 streaming loads; branch-free insert; LDS tree-merge.
// ---------------------------------------------------------------------------
__global__ __launch_bounds__(T1)
void ssd_top3_scan(const float* __restrict__ cls,
                   const int*   __restrict__ box_idxs,
                   float* __restrict__ wsv,
                   int*   __restrict__ wsi) {
    const int bx   = blockIdx.x;
    const int row  = bx >> 4;            // / SPLIT
    const int part = bx & (SPLIT - 1);
    const int t    = threadIdx.x;
    const int pos  = box_idxs[row];

    const v4f* p = (const v4f*)(cls + (size_t)row * NBOX + (size_t)part * SEG);
    const int base = part * SEG;

    float v0 = -INFINITY, v1 = -INFINITY, v2 = -INFINITY;
    int   i0 = 0x7fffffff, i1 = 0x7fffffff, i2 = 0x7fffffff;

#pragma unroll
    for (int it = 0; it < VPT; ++it) {
        const int vi = it * T1 + t;                       // coalesced float4 idx
        const v4f q = __builtin_nontemporal_load(p + vi); // b128, TH=NT
        const int gi = base + vi * 4;
        // Exclude the positive anchor via select (no divergence).
        const float x0 = (gi + 0 == pos) ? -INFINITY : q[0];
        const float x1 = (gi + 1 == pos) ? -INFINITY : q[1];
        const float x2 = (gi + 2 == pos) ? -INFINITY : q[2];
        const float x3 = (gi + 3 == pos) ? -INFINITY : q[3];
        ins3_inc(x0, gi + 0, v0, i0, v1, i1, v2, i2);
        ins3_inc(x1, gi + 1, v0, i0, v1, i1, v2, i2);
        ins3_inc(x2, gi + 2, v0, i0, v1, i1, v2, i2);
        ins3_inc(x3, gi + 3, v0, i0, v1, i1, v2, i2);
    }

    __shared__ float sv[T1 * 3];
    __shared__ int   si[T1 * 3];
    sv[t * 3 + 0] = v0; sv[t * 3 + 1] = v1; sv[t * 3 + 2] = v2;
    si[t * 3 + 0] = i0; si[t * 3 + 1] = i1; si[t * 3 + 2] = i2;
    __syncthreads();

    for (int s = T1 / 2; s > 0; s >>= 1) {
        if (t < s) {
            const int o = (t + s) * 3;
            ins3_full(sv[o + 0], si[o + 0], v0, i0, v1, i1, v2, i2);
            ins3_full(sv[o + 1], si[o + 1], v0, i0, v1, i1, v2, i2);
            ins3_full(sv[o + 2], si[o + 2], v0, i0, v1, i1, v2, i2);
            sv[t * 3 + 0] = v0; sv[t * 3 + 1] = v1; sv[t * 3 + 2] = v2;
            si[t * 3 + 0] = i0; si[t * 3 + 1] = i1; si[t * 3 + 2] = i2;
        }
        __syncthreads();
    }

    if (t == 0) {
        wsv[bx * 3 + 0] = v0; wsv[bx * 3 + 1] = v1; wsv[bx * 3 + 2] = v2;
        wsi[bx * 3 + 0] = i0; wsi[bx * 3 + 1] = i1; wsi[bx * 3 + 2] = i2;
    }
}

// ---------------------------------------------------------------------------
// Kernel 2: one wave32 per row. Shuffle-butterfly merge of the 16 segment
// triples, then compute the row's BCE + smooth-L1 partial.
// ---------------------------------------------------------------------------
__global__ __launch_bounds__(32)
void ssd_row_finalize(const float* __restrict__ cls,
                      const float* __restrict__ reg,
                      const int*   __restrict__ box_idxs,
                      const float* __restrict__ rt,
                      const float* __restrict__ wsv,
                      const int*   __restrict__ wsi,
                      float* __restrict__ ws2) {
    const int row  = blockIdx.x;
    const int lane = threadIdx.x;

    float v0 = -INFINITY, v1 = -INFINITY, v2 = -INFINITY;
    int   i0 = 0x7fffffff, i1 = 0x7fffffff, i2 = 0x7fffffff;
    if (lane < SPLIT) {
        const int o = (row * SPLIT + lane) * 3;
        v0 = wsv[o + 0]; v1 = wsv[o + 1]; v2 = wsv[o + 2];
        i0 = wsi[o + 0]; i1 = wsi[o + 1]; i2 = wsi[o + 2];
    }

    // Butterfly over lanes 0..15 (lanes >= SPLIT carry sentinels).
    for (int m = SPLIT / 2; m > 0; m >>= 1) {
        const float ov0 = __shfl_xor(v0, m), ov1 = __shfl_xor(v1, m), ov2 = __shfl_xor(v2, m);
        const int   oi0 = __shfl_xor(i0, m), oi1 = __shfl_xor(i1, m), oi2 = __shfl_xor(i2, m);
        ins3_full(ov0, oi0, v0, i0, v1, i1, v2, i2);
        ins3_full(ov1, oi1, v0, i0, v1, i1, v2, i2);
        ins3_full(ov2, oi2, v0, i0, v1, i1, v2, i2);
    }

    if (lane == 0) {
        const int pos = box_idxs[row];
        const float xp = cls[(size_t)row * NBOX + pos];
        float cl = bce(xp, 1.0f) + bce(v0, 0.0f) + bce(v1, 0.0f) + bce(v2, 0.0f);
        float rg = 0.0f;
#pragma unroll
        for (int c = 0; c < C; ++c) {
            const float pv = reg[((size_t)row * C + c) * NBOX + pos];
            const float tv = rt[row * C + c];
            const float d  = fabsf(pv - tv);
            rg += (d < 1.0f) ? 0.5f * d * d : d - 0.5f;
        }
        ws2[row] = cl + rg;   // both means are over B*4 = 512 elements
    }
}

// ---------------------------------------------------------------------------
// Kernel 3: sum the 128 row partials with the matrix pipe.
// With B = all-ones, D[i][j] = sum_k A[i][k]; summing column 0 of D
// (lane 0 VGPRs 0..7 plus lane 16 VGPRs 0..7) yields the sum of every
// A slot regardless of which ws2 element landed in which (M,K) slot.
// ---------------------------------------------------------------------------
__global__ __launch_bounds__(32)
void ssd_wmma_sum(const float* __restrict__ ws2, float* __restrict__ out) {
    const int lane = threadIdx.x;

    v2f a0, a1, ones;
    a0[0] = ws2[2 * lane + 0];  a0[1] = ws2[2 * lane + 1];        // partials 0..63
    a1[0] = ws2[64 + 2 * lane]; a1[1] = ws2[64 + 2 * lane + 1];   // partials 64..127
    ones[0] = 1.0f; ones[1] = 1.0f;

    v8f c = {};
    c = __builtin_amdgcn_wmma_f32_16x16x4_f32(false, a0, false, ones,
                                              (short)0, c, false, false);
    c = __builtin_amdgcn_wmma_f32_16x16x4_f32(false, a1, false, ones,
                                              (short)0, c, false, false);

    float s = c[0] + c[1] + c[2] + c[3] + c[4] + c[5] + c[6] + c[7];
    const float tot = __shfl(s, 0) + __shfl(s, 16);   // D[0..7][0] + D[8..15][0]
    if (lane == 0) out[0] = tot * (1.0f / 512.0f);
}

// ---------------------------------------------------------------------------
extern "C" void kernel_launch(void* const* d_in, const int* in_sizes, int n_in,
                              void* d_out, int out_size, void* d_ws, size_t ws_size,
                              hipStream_t stream) {
    const float* cls = (const float*)d_in[0];   // [B, NBOX]       f32
    const float* reg = (const float*)d_in[1];   // [B, C, NBOX]    f32
    const int*   idx = (const int*)  d_in[2];   // [B]             i32
    const float* rt  = (const float*)d_in[3];   // [B, C]          f32

    // Workspace layout (~49.5 KB total):
    //   wsv : B*SPLIT*3 floats  (segment top-3 values)
    //   wsi : B*SPLIT*3 ints    (segment top-3 indices)
    //   ws2 : B floats          (per-row loss partials)
    float* wsv = (float*)d_ws;
    int*   wsi = (int*)((char*)d_ws + (size_t)(B * SPLIT * 3) * sizeof(float));
    float* ws2 = (float*)((char*)d_ws + (size_t)(B * SPLIT * 3) * (sizeof(float) + sizeof(int)));

    ssd_top3_scan   <<<B * SPLIT, T1, 0, stream>>>(cls, idx, wsv, wsi);
    ssd_row_finalize<<<B,         32, 0, stream>>>(cls, reg, idx, rt, wsv, wsi, ws2);
    ssd_wmma_sum    <<<1,         32, 0, stream>>>(ws2, (float*)d_out);
}